// cumsum_with_decay_factor_12618613916100
// MI455X (gfx1250) — compile-verified
//
#include <hip/hip_runtime.h>
#include <stdint.h>

// Problem constants (reference: B,H,S,L = 2,12,2048,2048)
#define ROW_LEN 2048
#define SEG 64                         // elements per lane (32 lanes * 64 = 2048)
#define WPB 4                          // waves per block
#define BLOCK (WPB * 32)
#define CHUNKS_PER_ROW (ROW_LEN / (32 * 4))      // 16-byte chunks per lane = 16
#define PAD_WORDS 4                    // 4-float pad per 64 floats (keeps 16B align, kills bank conflicts)
#define ROW_WORDS_PADDED (ROW_LEN + (ROW_LEN / SEG) * PAD_WORDS)   // 2176 floats = 8704 B

// ---- CDNA5 async global<->LDS path (ASYNCcnt-tracked DMA) ----
#if defined(__has_builtin)
#if __has_builtin(__builtin_amdgcn_global_load_async_to_lds_b128) && \
    __has_builtin(__builtin_amdgcn_global_store_async_from_lds_b128)
#define USE_ASYNC_LDS 1
#endif
#endif
#ifndef USE_ASYNC_LDS
#define USE_ASYNC_LDS 0
#endif

typedef int v4i __attribute__((vector_size(16)));        // int __vector(4): global-side pointee
typedef __attribute__((address_space(3))) v4i as3_v4i;   // AS3-qualified int __vector(4): LDS-side pointee

static __device__ __forceinline__ void wait_asynccnt0() {
#if defined(__has_builtin) && __has_builtin(__builtin_amdgcn_s_wait_asynccnt)
  __builtin_amdgcn_s_wait_asynccnt(0);
#else
  asm volatile("s_wait_asynccnt 0" ::: "memory");
#endif
}

static __device__ __forceinline__ void wait_dscnt0() {
#if defined(__has_builtin) && __has_builtin(__builtin_amdgcn_s_wait_dscnt)
  __builtin_amdgcn_s_wait_dscnt(0);
#else
  asm volatile("s_wait_dscnt 0" ::: "memory");
#endif
}

// Padded LDS byte offset for global 16B-chunk index g (4 floats per chunk):
// word i -> i + 4*(i>>6); chunk g covers words 4g..4g+3, all in block (g>>4).
static __device__ __forceinline__ unsigned lds_chunk_byte(unsigned g) {
  return 16u * g + 16u * (g >> 4);
}

__global__ __launch_bounds__(BLOCK) void cumsum_decay_scan_kernel(
    const float* __restrict__ scores,
    const float* __restrict__ decay_w,
    float* __restrict__ out,
    int nrows) {
  __shared__ __align__(16) float lds[WPB * ROW_WORDS_PADDED];

  const int lane = threadIdx.x & 31;
  const int wav  = threadIdx.x >> 5;
  const int row  = blockIdx.x * WPB + wav;
  if (row >= nrows) return;

  float* slice = &lds[wav * ROW_WORDS_PADDED];

  const float aw = fabsf(decay_w[0]);
  const float r  = __expf(-aw);                 // per-step decay
  const float R  = __expf(-aw * (float)SEG);    // r^64, per-segment decay

  const char* grow = (const char*)(scores + (size_t)row * ROW_LEN);

  // ---- Stage the row into LDS (padded layout), globally coalesced ----
#if USE_ASYNC_LDS
#pragma unroll
  for (int j = 0; j < CHUNKS_PER_ROW; ++j) {
    unsigned g = (unsigned)(j * 32 + lane);   // 16B chunk id within the row
    __builtin_amdgcn_global_load_async_to_lds_b128(
        (v4i*)(grow + 16u * g),
        (as3_v4i*)((char*)slice + lds_chunk_byte(g)),
        0, 0);
  }
  asm volatile("" ::: "memory");
  wait_asynccnt0();   // LDS now holds the row
#else
#pragma unroll
  for (int j = 0; j < CHUNKS_PER_ROW; ++j) {
    unsigned g = (unsigned)(j * 32 + lane);
    float4 t = *(const float4*)(grow + 16u * g);
    *(float4*)((char*)slice + lds_chunk_byte(g)) = t;
  }
  // same-wave DS ops are in-order: subsequent ds_loads see these stores
#endif

  // ---- Per-lane local scan over its contiguous 64-element segment ----
  // Element 64k+t lives at padded word 68k+t -> lane base 272k bytes (16B aligned)
  const unsigned lbase = 68u * (unsigned)lane;   // word offset within slice
  float4 v[16];
#pragma unroll
  for (int j = 0; j < 16; ++j)
    v[j] = *(const float4*)(slice + lbase + 4u * j);

  float acc = 0.f;
#pragma unroll
  for (int j = 0; j < 16; ++j) {
    v[j].x = fmaf(r, acc,    v[j].x);
    v[j].y = fmaf(r, v[j].x, v[j].y);
    v[j].z = fmaf(r, v[j].y, v[j].z);
    v[j].w = fmaf(r, v[j].z, v[j].w);
    acc = v[j].w;
  }

  // ---- Cross-lane scan of affine maps f_k(x) = R*x + C_k (Hillis-Steele) ----
  float A = R, Bv = acc;
#pragma unroll
  for (int d = 1; d < 32; d <<= 1) {
    float Au = __shfl_up(A,  (unsigned)d, 32);
    float Bu = __shfl_up(Bv, (unsigned)d, 32);
    if (lane >= d) {
      Bv = fmaf(A, Bu, Bv);   // compose: (A,B) o (Au,Bu) = (A*Au, A*Bu + B)
      A  = A * Au;
    }
  }
  float T = __shfl_up(Bv, 1u, 32);   // carry entering this lane's segment
  if (lane == 0) T = 0.f;

  // ---- Fixup: out[64k+t] = local[t] + r^{t+1} * T ----
  const float r2 = r * r, r3 = r2 * r, r4 = r2 * r2;
  float base = r * T;   // r^{4j+1} * T for chunk j
#pragma unroll
  for (int j = 0; j < 16; ++j) {
    v[j].x += base;
    v[j].y = fmaf(base, r,  v[j].y);
    v[j].z = fmaf(base, r2, v[j].z);
    v[j].w = fmaf(base, r3, v[j].w);
    base *= r4;
  }

  // ---- Results back to LDS (padded), then coalesced store to global ----
#pragma unroll
  for (int j = 0; j < 16; ++j)
    *(float4*)(slice + lbase + 4u * j) = v[j];

  asm volatile("" ::: "memory");
  char* orow = (char*)(out + (size_t)row * ROW_LEN);
#if USE_ASYNC_LDS
  wait_dscnt0();   // async DMA engine is not ordered with the DS unit
#pragma unroll
  for (int j = 0; j < CHUNKS_PER_ROW; ++j) {
    unsigned g = (unsigned)(j * 32 + lane);
    __builtin_amdgcn_global_store_async_from_lds_b128(
        (v4i*)(orow + 16u * g),
        (as3_v4i*)((char*)slice + lds_chunk_byte(g)),
        0, 0);
  }
  wait_asynccnt0();  // don't release LDS while DMA reads it
#else
#pragma unroll
  for (int j = 0; j < CHUNKS_PER_ROW; ++j) {
    unsigned g = (unsigned)(j * 32 + lane);
    float4 t = *(const float4*)((char*)slice + lds_chunk_byte(g));
    *(float4*)(orow + 16u * g) = t;
  }
#endif
}

extern "C" void kernel_launch(void* const* d_in, const int* in_sizes, int n_in,
                              void* d_out, int out_size, void* d_ws, size_t ws_size,
                              hipStream_t stream) {
  (void)n_in; (void)out_size; (void)d_ws; (void)ws_size;
  const float* scores  = (const float*)d_in[0];
  const float* decay_w = (const float*)d_in[1];
  float* out = (float*)d_out;

  const int nrows  = in_sizes[0] / ROW_LEN;          // B*H*S = 49152
  const int blocks = (nrows + WPB - 1) / WPB;        // 12288 blocks of 128 threads
  cumsum_decay_scan_kernel<<<blocks, BLOCK, 0, stream>>>(scores, decay_w, out, nrows);
}